// SingleColumnModel_27496380629250
// MI455X (gfx1250) — compile-verified
//
#include <hip/hip_runtime.h>
#include <stdint.h>

// ---------------------------------------------------------------------------
// Single-column ocean model (KPP-style) for MI455X / gfx1250.
// One lane = one column (wave32). Global<->LDS movement via the CDNA5
// Tensor Data Mover (TENSOR_LOAD_TO_LDS / TENSOR_STORE_FROM_LDS), double-
// buffered with s_wait_tensorcnt<=1 so one tile DMA is always in flight.
// D# pad feature gives a conflict-free (pitch-33) transposed LDS tile on load.
// ---------------------------------------------------------------------------

#define COLS   128          // columns (threads) per block
#define CHUNK  32           // z-levels staged per TDM tile
#define NZMAX  132
#define LDSPITCH (CHUNK + 1)

#define DT_C       50.0f
#define FCOR_C     1.0e-4f
#define USTR_SFC_C 1.0e-4f
#define VSTR_SFC_C 0.0f
#define TFLX_SFC_C (-1.0e-5f)
#define RFLX_C     (150.0f / (1024.0f * 3985.0f))
#define CP_C       3985.0f
#define ALPHA_C    2.0e-4f
#define GRAV_C     9.81f
#define GAMMA_C    0.55f
#define MU1_C      0.35f
#define MU2_C      23.0f
#define R1_C       0.58f

typedef __attribute__((ext_vector_type(4))) unsigned int uint4v;
typedef __attribute__((ext_vector_type(8))) int          int8v;
typedef __attribute__((ext_vector_type(4))) int          int4v;

// Tensor-DMA descriptor (D#) for a 2D fp32 tile:
//   tensor : [nrows_total x rowlen] row-major, row stride = rowlen elems
//   tile   : [tile_rows x tile_z] starting at gptr
//   pad    : loads insert 1 DWORD after every 32 DWORDs in LDS (pitch 33)
__device__ __forceinline__ void tdm_tile(const void* gptr, uint32_t lds_off,
                                         uint32_t rowlen, uint32_t nrows_total,
                                         uint32_t tile_z, uint32_t tile_rows,
                                         bool pad, bool is_store) {
  uint64_t ga = (uint64_t)(uintptr_t)gptr;
  uint4v g0;
  g0[0] = 1u;                                   // count=1, user descriptor
  g0[1] = lds_off;                              // lds_addr (bytes)
  g0[2] = (uint32_t)ga;                         // global_addr[31:0]
  g0[3] = (uint32_t)((ga >> 32) & 0x1FFFFFFull) // global_addr[56:32]
        | (2u << 30);                           // type = 2 ("image")

  uint32_t d0 = (2u << 16);                     // data_size = 4 bytes
  if (pad) d0 |= (1u << 20) | (4u << 22);       // pad_enable, 32DW interval, 1DW
  int8v g1;
  g1[0] = (int)d0;
  g1[1] = (int)((rowlen & 0xFFFFu) << 16);                          // tensor_dim0 lo16
  g1[2] = (int)((rowlen >> 16) | ((nrows_total & 0xFFFFu) << 16));  // dim0 hi / dim1 lo
  g1[3] = (int)((nrows_total >> 16) | (tile_z << 16));              // dim1 hi / tile_dim0
  g1[4] = (int)tile_rows;                                           // tile_dim1 (tile_dim2=0)
  g1[5] = (int)rowlen;                                              // tensor_dim0_stride[31:0]
  g1[6] = 0;
  g1[7] = 0;
  int4v g2 = {0, 0, 0, 0};                      // 2D tile: groups 2/3 unused
  int4v g3 = {0, 0, 0, 0};
  int8v g4 = {0, 0, 0, 0, 0, 0, 0, 0};          // extra group (clang-23 form)
  if (is_store)
    __builtin_amdgcn_tensor_store_from_lds(g0, g1, g2, g3, g4, 0);
  else
    __builtin_amdgcn_tensor_load_to_lds(g0, g1, g2, g3, g4, 0);
}

// Thomas solve for two RHS sharing one coefficient set; coefficients are
// regenerated on the fly (inputs are step-invariant), only cp is stored.
__device__ __forceinline__ void solve2(const float* ak, const float* hz,
                                       float* f1, float* f2, float* cp, int nz) {
  float cpp = 0.0f, d1 = 0.0f, d2 = 0.0f;
  for (int i = 0; i < nz; ++i) {
    float ai = (i > 0)      ? (-2.0f * DT_C * ak[i]     / (hz[i - 1] + hz[i])) : 0.0f;
    float ci = (i < nz - 1) ? (-2.0f * DT_C * ak[i + 1] / (hz[i] + hz[i + 1])) : 0.0f;
    float bi = hz[i] - ai - ci;
    float inv = 1.0f / (bi - ai * cpp);
    cpp = ci * inv;
    cp[i] = cpp;
    d1 = (f1[i] - ai * d1) * inv; f1[i] = d1;
    d2 = (f2[i] - ai * d2) * inv; f2[i] = d2;
  }
  float x1 = 0.0f, x2 = 0.0f;
  for (int i = nz - 1; i >= 0; --i) {
    x1 = f1[i] - cp[i] * x1; f1[i] = x1;
    x2 = f2[i] - cp[i] * x2; f2[i] = x2;
  }
}

__global__ __launch_bounds__(COLS)
void scm_kernel(const float* __restrict__ u0, const float* __restrict__ v0,
                const float* __restrict__ t0, const float* __restrict__ s0,
                const float* __restrict__ akv_g, const float* __restrict__ akt_g,
                const float* __restrict__ eps_g, const float* __restrict__ hz_g,
                const float* __restrict__ zw_g, const int* __restrict__ nsteps_p,
                float* __restrict__ out, int Bn, int nz) {
  __shared__ float smem[2 * COLS * LDSPITCH];   // two tile buffers (padded)

  const int tid  = (int)threadIdx.x;
  const int col0 = (int)blockIdx.x * COLS;
  const int col  = col0 + tid;
  const int colc = col < Bn ? col : Bn - 1;     // clamp for direct loads
  const int nzp1 = nz + 1;
  const uint32_t lds_base = (uint32_t)(uintptr_t)(void*)smem;

  // Per-lane column state in scratch (swizzled -> same-k access coalesces).
  float u[NZMAX], v[NZMAX], t[NZMAX], s[NZMAX];
  float hz[NZMAX], srcT[NZMAX], cp[NZMAX];
  float akv[NZMAX + 1], akt[NZMAX + 1], sw[NZMAX + 1];

  const int zchunks = nz / CHUNK;               // 4 for nz=128
  const int ntile   = 4 * zchunks;              // u,v,t,s

  // ---- Stage u,v,t,s via double-buffered TDM loads --------------------------
  {
    const float* srcs[4] = {u0, v0, t0, s0};
    float*       dsts[4] = {u, v, t, s};
    if (tid < 32)                                // prologue: tile 0 -> buf 0
      tdm_tile(srcs[0] + (size_t)col0 * nz, lds_base,
               (uint32_t)nz, (uint32_t)Bn, CHUNK, COLS, true, false);
    for (int m = 0; m < ntile; ++m) {
      const int a = m / zchunks, c = m - a * zchunks;
      if (m + 1 < ntile) {
        const int a2 = (m + 1) / zchunks, c2 = (m + 1) - a2 * zchunks;
        if (tid < 32) {
          tdm_tile(srcs[a2] + (size_t)col0 * nz + (size_t)c2 * CHUNK,
                   lds_base + (uint32_t)(((m + 1) & 1) * COLS * LDSPITCH * 4),
                   (uint32_t)nz, (uint32_t)Bn, CHUNK, COLS, true, false);
          __builtin_amdgcn_s_wait_tensorcnt(1);  // tile m resident (in-order)
        }
      } else if (tid < 32) {
        __builtin_amdgcn_s_wait_tensorcnt(0);    // last tile resident
      }
      __syncthreads();                           // tile m visible to all waves
      const float* buf = smem + (m & 1) * COLS * LDSPITCH;
      for (int k = 0; k < CHUNK; ++k)            // pitch 33: conflict-free
        dsts[a][c * CHUNK + k] = buf[tid * LDSPITCH + k];
      __syncthreads();                           // done before buf is re-filled
    }
  }

  // ---- Direct per-lane loads (setup-only / small arrays) ----
  for (int k = 0; k < nz; ++k)    hz[k]  = hz_g[(size_t)colc * nz + k];
  for (int k = 0; k < nzp1; ++k) { akv[k] = akv_g[(size_t)colc * nzp1 + k];
                                   akt[k] = akt_g[(size_t)colc * nzp1 + k]; }

  // ---- swr_frac (reversed cumprod from surface) ----
  {
    float p1 = 1.0f, p2 = 1.0f;
    for (int k = nz - 1; k >= 0; --k) {
      float x1 = -hz[k] * (1.0f / MU1_C);
      float x2 = -hz[k] * (1.0f / MU2_C);
      float f1 = (x1 > -20.0f) ? __expf(x1) : 0.0f;
      float f2 = (x2 > -20.0f) ? __expf(x2) : 0.0f;
      p1 *= f1; p2 *= f2;
      sw[k] = R1_C * p1 + (1.0f - R1_C) * p2;
    }
    sw[nz] = 1.0f;
  }

  // ---- Step-invariant temperature source term ----
  for (int k = 0; k < nz; ++k) {
    float fcl = (k == 0) ? 0.0f : RFLX_C * sw[k];
    float fch = RFLX_C * sw[k + 1] + ((k == nz - 1) ? TFLX_SFC_C : 0.0f);
    float zl = zw_g[(size_t)colc * nzp1 + k];
    float zh = zw_g[(size_t)colc * nzp1 + k + 1];
    float el = eps_g[(size_t)colc * nzp1 + k];
    float eh = eps_g[(size_t)colc * nzp1 + k + 1];
    float cffm = el / (CP_C - ALPHA_C * GRAV_C * zl);
    float cffp = eh / (CP_C - ALPHA_C * GRAV_C * zh);
    srcT[k] = DT_C * (fch - fcl) + DT_C * 0.5f * hz[k] * (cffp + cffm);
  }

  // ---- Time stepping ----
  const int nsteps = nsteps_p[0];
  const float cff  = (DT_C * FCOR_C) * (DT_C * FCOR_C);
  const float cff1 = 1.0f / (1.0f + GAMMA_C * GAMMA_C * cff);
  const float aco  = 1.0f - GAMMA_C * (1.0f - GAMMA_C) * cff;
  for (int st = 0; st < nsteps; ++st) {
    for (int k = 0; k < nz; ++k) {               // tracers
      t[k] = hz[k] * t[k] + srcT[k];
      s[k] = hz[k] * s[k];
    }
    solve2(akt, hz, t, s, cp, nz);
    for (int k = 0; k < nz; ++k) {               // dynamics (ref: v uses new u)
      float un = cff1 * hz[k] * (aco * u[k] + DT_C * FCOR_C * v[k]);
      float vn = cff1 * hz[k] * (aco * v[k] - DT_C * FCOR_C * un);
      u[k] = un; v[k] = vn;
    }
    u[nz - 1] += DT_C * USTR_SFC_C;
    v[nz - 1] += DT_C * VSTR_SFC_C;
    solve2(akv, hz, u, v, cp, nz);
  }

  // ---- Write back via double-buffered TDM stores (unpadded, pitch 32).
  // Per-lane rotation ((33c+j) mod 64 distinct) keeps LDS writes conflict-free.
  {
    const float* outs[4] = {u, v, t, s};
    __syncthreads();                             // load-phase LDS fully consumed
    for (int m = 0; m < ntile; ++m) {
      const int a = m / zchunks, c = m - a * zchunks;
      float* buf = smem + (m & 1) * COLS * CHUNK;
      for (int j = 0; j < CHUNK; ++j) {
        int k = (tid + j) & (CHUNK - 1);
        buf[tid * CHUNK + k] = outs[a][c * CHUNK + k];
      }
      __syncthreads();                           // tile m complete in LDS
      if (tid < 32) {
        tdm_tile(out + (size_t)a * Bn * nz + (size_t)col0 * nz + (size_t)c * CHUNK,
                 lds_base + (uint32_t)((m & 1) * COLS * CHUNK * 4),
                 (uint32_t)nz, (uint32_t)Bn, CHUNK, COLS, false, true);
        __builtin_amdgcn_s_wait_tensorcnt(1);    // store m-1 drained -> other buf free
      }
      __syncthreads();                           // gate next buffer overwrite
    }
    if (tid < 32) __builtin_amdgcn_s_wait_tensorcnt(0);  // drain pipe
  }
}

extern "C" void kernel_launch(void* const* d_in, const int* in_sizes, int n_in,
                              void* d_out, int out_size, void* d_ws, size_t ws_size,
                              hipStream_t stream) {
  (void)n_in; (void)d_ws; (void)ws_size; (void)out_size;
  const float* u   = (const float*)d_in[0];
  const float* v   = (const float*)d_in[1];
  const float* t   = (const float*)d_in[2];
  const float* s   = (const float*)d_in[3];
  const float* akv = (const float*)d_in[4];
  const float* akt = (const float*)d_in[5];
  const float* eps = (const float*)d_in[6];
  const float* hz  = (const float*)d_in[7];
  const float* zw  = (const float*)d_in[8];
  const int*   nst = (const int*)d_in[9];
  float* out = (float*)d_out;

  const int nz = 128;
  const int B  = in_sizes[0] / nz;           // 32768
  dim3 grid((B + COLS - 1) / COLS), block(COLS);
  scm_kernel<<<grid, block, 0, stream>>>(u, v, t, s, akv, akt, eps, hz, zw,
                                         nst, out, B, nz);
}